// PieceGNN_6055903888072
// MI455X (gfx1250) — compile-verified
//
#include <hip/hip_runtime.h>
#include <hip/hip_bf16.h>
#include <math.h>

#define DIN 128
#define DH  128

typedef float v2f __attribute__((ext_vector_type(2)));
typedef float v8f __attribute__((ext_vector_type(8)));

// ---------------------------------------------------------------- degree/norm

__global__ void deg_init_kernel(float* __restrict__ deg, int n) {
    int i = blockIdx.x * blockDim.x + threadIdx.x;
    if (i < n) deg[i] = 1.0f;   // self-loop contributes 1 to every node
}

__global__ void deg_accum_kernel(const long long* __restrict__ dst,
                                 float* __restrict__ deg, long long e) {
    long long i = (long long)blockIdx.x * blockDim.x + threadIdx.x;
    if (i < e) atomicAdd(&deg[(int)dst[i]], 1.0f);
}

__global__ void deg_rsqrt_kernel(float* __restrict__ deg, int n) {
    int i = blockIdx.x * blockDim.x + threadIdx.x;
    if (i < n) deg[i] = rsqrtf(deg[i]);   // deg >= 1 always (self-loops)
}

__global__ void norm_kernel(const long long* __restrict__ src,
                            const long long* __restrict__ dst,
                            const float* __restrict__ dinv,
                            float* __restrict__ norm, long long e) {
    long long i = (long long)blockIdx.x * blockDim.x + threadIdx.x;
    if (i < e) norm[i] = dinv[(int)src[i]] * dinv[(int)dst[i]];
}

// ---------------------------------------------------------------- WMMA GEMM
// H[nrows x 128] = X[nrows x 128] * W[128 x 128], fp32 via V_WMMA_F32_16X16X4_F32.
// Block = 128 threads (4 waves). Block owns 16 output rows; each wave owns two
// 16x16 column tiles. A tile staged in LDS (stride 132 -> conflict-free frag reads).

__global__ void __launch_bounds__(128)
gemm_wmma_kernel(const float* __restrict__ X, const float* __restrict__ W,
                 float* __restrict__ H, int nrows) {
    __shared__ float At[16 * 132];
    const int r0 = blockIdx.x * 16;

    // Cooperative load of A tile: 16 rows x 128 cols = 512 float4
    for (int j = threadIdx.x; j < 512; j += 128) {
        const int row = j >> 5;           // 0..15
        const int c4  = (j & 31) << 2;    // 0..124 step 4
        float4 v = make_float4(0.f, 0.f, 0.f, 0.f);
        if (r0 + row < nrows)
            v = *(const float4*)(X + (size_t)(r0 + row) * DIN + c4);
        *(float4*)(&At[row * 132 + c4]) = v;
    }
    __syncthreads();

    const int lane = threadIdx.x & 31;
    const int wave = threadIdx.x >> 5;
    const int half = lane >> 4;           // 0: lanes 0-15, 1: lanes 16-31
    const int lrow = lane & 15;
    const int c0   = wave * 32;           // this wave's first output column

    v8f acc0 = {0.f, 0.f, 0.f, 0.f, 0.f, 0.f, 0.f, 0.f};
    v8f acc1 = {0.f, 0.f, 0.f, 0.f, 0.f, 0.f, 0.f, 0.f};

    for (int kk = 0; kk < DIN; kk += 4) {
        const int krow = kk + 2 * half;   // A layout: v0 -> K=0/2, v1 -> K=1/3
        v2f a;
        a.x = At[lrow * 132 + krow];
        a.y = At[lrow * 132 + krow + 1];

        const float* wp = W + (size_t)krow * DH + c0 + lrow;
        v2f b0, b1;
        b0.x = wp[0];
        b0.y = wp[DH];
        b1.x = wp[16];
        b1.y = wp[DH + 16];

        if (kk + 4 < DIN)   // pull next K-slab of W toward the WGP
            __builtin_prefetch(W + (size_t)(krow + 4) * DH + c0 + lrow, 0, 1);

        acc0 = __builtin_amdgcn_wmma_f32_16x16x4_f32(
            false, a, false, b0, (short)0, acc0, false, false);
        acc1 = __builtin_amdgcn_wmma_f32_16x16x4_f32(
            false, a, false, b1, (short)0, acc1, false, false);
    }

    // C/D layout: VGPR v -> row (v + 8*half), col = lrow
    for (int v = 0; v < 8; ++v) {
        const int row = r0 + v + 8 * half;
        if (row < nrows) {
            H[(size_t)row * DH + c0 + lrow]      = acc0[v];
            H[(size_t)row * DH + c0 + 16 + lrow] = acc1[v];
        }
    }
}

// ---------------------------------------------------------------- scatter-add
// One wave32 per edge slot; lane handles 4 consecutive floats (512B/edge,
// fully coalesced; h and the accumulator stay L2-resident at 51MB each).

__global__ void __launch_bounds__(256)
scatter_kernel(const float* __restrict__ h, const long long* __restrict__ src,
               const long long* __restrict__ dst, const float* __restrict__ norm,
               const float* __restrict__ dinv, float* __restrict__ out,
               long long e, int n) {
    const long long gid = (long long)blockIdx.x * blockDim.x + threadIdx.x;
    const long long eid = gid >> 5;
    const int lane = (int)(gid & 31);
    if (eid >= e + n) return;

    int s, d; float w;
    if (eid < e) {
        s = (int)src[eid]; d = (int)dst[eid]; w = norm[eid];
    } else {                      // synthesized self-loop edge
        s = d = (int)(eid - e);
        const float di = dinv[s];
        w = di * di;
    }

    const float4 hv = *(const float4*)(h + (size_t)s * DH + lane * 4);
    float* o = out + (size_t)d * DH + lane * 4;
    atomicAdd(o + 0, w * hv.x);
    atomicAdd(o + 1, w * hv.y);
    atomicAdd(o + 2, w * hv.z);
    atomicAdd(o + 3, w * hv.w);
}

// ---------------------------------------------------------------- bias + GELU

__global__ void bias_act_kernel(float* __restrict__ y, const float* __restrict__ b,
                                int total, int gelu) {
    const int i = blockIdx.x * blockDim.x + threadIdx.x;
    if (i >= total) return;
    float x = y[i] + b[i & (DH - 1)];
    if (gelu) x = 0.5f * x * (1.0f + erff(x * 0.70710678118654752f));
    y[i] = x;
}

// ---------------------------------------------------------------- launch

extern "C" void kernel_launch(void* const* d_in, const int* in_sizes, int n_in,
                              void* d_out, int out_size, void* d_ws, size_t ws_size,
                              hipStream_t stream) {
    const float*     x   = (const float*)d_in[0];
    const long long* ei  = (const long long*)d_in[1];   // int64 (2, E)
    const float*     W1  = (const float*)d_in[2];
    const float*     b1  = (const float*)d_in[3];
    const float*     W2  = (const float*)d_in[4];
    const float*     b2  = (const float*)d_in[5];
    const float*     W3  = (const float*)d_in[6];
    const float*     b3  = (const float*)d_in[7];

    const int n = in_sizes[0] / DIN;
    const long long e = (long long)in_sizes[1] / 2;
    const long long* src  = ei;
    const long long* dstp = ei + e;

    float* ws = (float*)d_ws;
    size_t off = 0;
    auto alloc = [&](size_t cnt) {
        float* p = ws + off;
        off += (cnt + 63) & ~(size_t)63;
        return p;
    };
    float* dinv = alloc((size_t)n);        // degree, then rsqrt in place
    float* norm = alloc((size_t)e);
    float* h    = alloc((size_t)n * DH);   // GEMM output
    float* acc  = alloc((size_t)n * DH);   // scatter accumulator / next input
    float* outp = (float*)d_out;

    const int T = 256;
    const int nb = (n + T - 1) / T;
    const int eb = (int)((e + T - 1) / T);
    deg_init_kernel <<<nb, T, 0, stream>>>(dinv, n);
    deg_accum_kernel<<<eb, T, 0, stream>>>(dstp, dinv, e);
    deg_rsqrt_kernel<<<nb, T, 0, stream>>>(dinv, n);
    norm_kernel     <<<eb, T, 0, stream>>>(src, dstp, dinv, norm, e);

    const int gemm_blocks = (n + 15) / 16;
    const long long slots = e + n;
    const int scat_blocks = (int)((slots * 32 + T - 1) / T);
    const int total = n * DH;
    const int ba_blocks = (total + T - 1) / T;
    const size_t bytes = (size_t)n * DH * sizeof(float);

    // Layer 1: h = X@W1; acc = scatter(h); acc = gelu(acc + b1)
    gemm_wmma_kernel<<<gemm_blocks, 128, 0, stream>>>(x, W1, h, n);
    hipMemsetAsync(acc, 0, bytes, stream);
    scatter_kernel  <<<scat_blocks, T, 0, stream>>>(h, src, dstp, norm, dinv, acc, e, n);
    bias_act_kernel <<<ba_blocks, T, 0, stream>>>(acc, b1, total, 1);

    // Layer 2
    gemm_wmma_kernel<<<gemm_blocks, 128, 0, stream>>>(acc, W2, h, n);
    hipMemsetAsync(acc, 0, bytes, stream);
    scatter_kernel  <<<scat_blocks, T, 0, stream>>>(h, src, dstp, norm, dinv, acc, e, n);
    bias_act_kernel <<<ba_blocks, T, 0, stream>>>(acc, b2, total, 1);

    // Layer 3 (no GELU)
    gemm_wmma_kernel<<<gemm_blocks, 128, 0, stream>>>(acc, W3, h, n);
    hipMemsetAsync(outp, 0, bytes, stream);
    scatter_kernel  <<<scat_blocks, T, 0, stream>>>(h, src, dstp, norm, dinv, outp, e, n);
    bias_act_kernel <<<ba_blocks, T, 0, stream>>>(outp, b3, total, 0);

    (void)n_in; (void)out_size; (void)ws_size;
}